// _HandwrittenGATConv_1606317769044
// MI455X (gfx1250) — compile-verified
//
#include <hip/hip_runtime.h>
#include <hip/hip_bf16.h>

#define N_NODES 50000
#define N_EDGES 800000
#define IN_DIM  128
#define OUT_DIM 64

typedef __attribute__((ext_vector_type(2))) float v2f;
typedef __attribute__((ext_vector_type(8))) float v8f;

// ---------------------------------------------------------------------------
// Workspace layout (bytes, all 16B aligned):
//   h      : [0,            12,800,000)   50000*64 f32
//   e_arr  : [12,800,000,   16,000,000)   800000 f32
//   a_src  : [16,000,000,   16,200,000)   50000 f32
//   a_dst  : [16,200,000,   16,400,000)   50000 f32
//   m      : [16,400,000,   16,600,000)   50000 f32
//   denom  : [16,600,000,   16,800,000)   50000 f32
// ---------------------------------------------------------------------------

__global__ void init_kernel(float* m, float* denom, float* out) {
    int i = blockIdx.x * blockDim.x + threadIdx.x;
    if (i < N_NODES) {
        m[i]     = -__builtin_inff();
        denom[i] = 0.0f;
    }
    if (i < N_NODES * OUT_DIM) out[i] = 0.0f;
}

// h = x @ W via V_WMMA_F32_16X16X4_F32.  One wave handles a 16-row tile
// across all 64 output columns (4 accumulators).  W staged in LDS.
__global__ void gemm_wmma_kernel(const float* __restrict__ x,
                                 const float* __restrict__ W,
                                 float* __restrict__ h) {
    __shared__ float sW[IN_DIM * OUT_DIM]; // 32 KB
    const int tid = threadIdx.x;
    for (int i = tid; i < IN_DIM * OUT_DIM; i += blockDim.x) sW[i] = W[i];
    __syncthreads();

    const int wave = tid >> 5;
    const int lane = tid & 31;
    const int half = lane >> 4;   // 0: lanes 0-15, 1: lanes 16-31
    const int l16  = lane & 15;

    const int tile = blockIdx.x * 4 + wave;   // 16-row tile index (wave-uniform)
    if (tile * 16 >= N_NODES) return;
    const int r0 = tile * 16;

    v8f acc0 = {}, acc1 = {}, acc2 = {}, acc3 = {};

    // A 16x4 f32 layout: lanes 0-15 hold K=0,1 (VGPR0,1); lanes 16-31 hold K=2,3
    const float* xrow = x + (size_t)(r0 + l16) * IN_DIM + 2 * half;

    for (int k = 0; k < IN_DIM; k += 4) {
        v2f a;
        a.x = xrow[k];
        a.y = xrow[k + 1];

        // B 4x16 f32 layout (mirrors A): lanes 0-15 hold K=0,1; lanes 16-31 K=2,3
        const float* w0 = sW + (k + 2 * half) * OUT_DIM + l16;
        const float* w1 = w0 + OUT_DIM;
        v2f b0; b0.x = w0[0];  b0.y = w1[0];
        v2f b1; b1.x = w0[16]; b1.y = w1[16];
        v2f b2; b2.x = w0[32]; b2.y = w1[32];
        v2f b3; b3.x = w0[48]; b3.y = w1[48];

        acc0 = __builtin_amdgcn_wmma_f32_16x16x4_f32(false, a, false, b0, (short)0, acc0, false, false);
        acc1 = __builtin_amdgcn_wmma_f32_16x16x4_f32(false, a, false, b1, (short)0, acc1, false, false);
        acc2 = __builtin_amdgcn_wmma_f32_16x16x4_f32(false, a, false, b2, (short)0, acc2, false, false);
        acc3 = __builtin_amdgcn_wmma_f32_16x16x4_f32(false, a, false, b3, (short)0, acc3, false, false);
    }

    // C/D layout: VGPR r -> row (r + 8*half), col = l16 + 16*t
    float* hbase = h + (size_t)r0 * OUT_DIM;
#pragma unroll
    for (int r = 0; r < 8; ++r) {
        const int row = r + 8 * half;
        float* hr = hbase + row * OUT_DIM + l16;
        hr[0]  = acc0[r];
        hr[16] = acc1[r];
        hr[32] = acc2[r];
        hr[48] = acc3[r];
    }
}

// Per-node attention scalars: a_src[n] = h[n] . att[0:64], a_dst[n] = h[n] . att[64:128]
__global__ void node_att_kernel(const float* __restrict__ h,
                                const float* __restrict__ att,
                                float* __restrict__ a_src,
                                float* __restrict__ a_dst) {
    const int gwave = (blockIdx.x * blockDim.x + threadIdx.x) >> 5;
    const int lane  = threadIdx.x & 31;
    if (gwave >= N_NODES) return;
    const float* hr = h + (size_t)gwave * OUT_DIM;
    const float h0 = hr[lane];
    const float h1 = hr[lane + 32];
    float ps = h0 * att[lane]      + h1 * att[lane + 32];
    float pd = h0 * att[64 + lane] + h1 * att[96 + lane];
#pragma unroll
    for (int off = 16; off; off >>= 1) {
        ps += __shfl_xor(ps, off, 32);
        pd += __shfl_xor(pd, off, 32);
    }
    if (lane == 0) {
        a_src[gwave] = ps;
        a_dst[gwave] = pd;
    }
}

// Float atomic max via sign-aware int punning (monotone under mixed updates)
__device__ __forceinline__ void atomicMaxFloat(float* addr, float val) {
    const int iv = __float_as_int(val);
    if (iv >= 0) atomicMax((int*)addr, iv);
    else         atomicMin((unsigned int*)addr, (unsigned int)iv);
}

__global__ void edge_logit_kernel(const int* __restrict__ src,
                                  const int* __restrict__ dst,
                                  const float* __restrict__ a_src,
                                  const float* __restrict__ a_dst,
                                  float* __restrict__ e_arr,
                                  float* __restrict__ m) {
    const int e = blockIdx.x * blockDim.x + threadIdx.x;
    if (e >= N_EDGES) return;
    const int s = src[e];
    const int d = dst[e];
    float v = a_src[s] + a_dst[d];
    v = v > 0.0f ? v : 0.2f * v;          // leaky_relu(0.2)
    e_arr[e] = v;
    atomicMaxFloat(&m[d], v);
}

__global__ void edge_exp_kernel(const int* __restrict__ dst,
                                const float* __restrict__ m,
                                float* __restrict__ e_arr,
                                float* __restrict__ denom) {
    const int e = blockIdx.x * blockDim.x + threadIdx.x;
    if (e >= N_EDGES) return;
    const int d = dst[e];
    const float ex = __expf(e_arr[e] - m[d]);
    e_arr[e] = ex;
    atomicAdd(&denom[d], ex);
}

// 16 threads per edge, float4 gather of h[src], 4 atomic adds into agg (d_out)
__global__ void edge_aggregate_kernel(const int* __restrict__ src,
                                      const int* __restrict__ dst,
                                      const float* __restrict__ e_arr,
                                      const float* __restrict__ denom,
                                      const float* __restrict__ h,
                                      float* __restrict__ agg) {
    const int t  = blockIdx.x * blockDim.x + threadIdx.x;
    const int e  = t >> 4;
    const int c4 = (t & 15) * 4;
    if (e >= N_EDGES) return;
    const int s = src[e];
    const int d = dst[e];
    const float alpha = e_arr[e] / denom[d];
    const float4 hv = *(const float4*)(h + (size_t)s * OUT_DIM + c4);
    float* out = agg + (size_t)d * OUT_DIM + c4;
    atomicAdd(out + 0, alpha * hv.x);
    atomicAdd(out + 1, alpha * hv.y);
    atomicAdd(out + 2, alpha * hv.z);
    atomicAdd(out + 3, alpha * hv.w);
}

__global__ void elu_kernel(float* __restrict__ out) {
    const int i = blockIdx.x * blockDim.x + threadIdx.x;
    if (i >= (N_NODES * OUT_DIM) / 4) return;
    float4 v = ((float4*)out)[i];
    v.x = v.x > 0.0f ? v.x : __expf(v.x) - 1.0f;
    v.y = v.y > 0.0f ? v.y : __expf(v.y) - 1.0f;
    v.z = v.z > 0.0f ? v.z : __expf(v.z) - 1.0f;
    v.w = v.w > 0.0f ? v.w : __expf(v.w) - 1.0f;
    ((float4*)out)[i] = v;
}

extern "C" void kernel_launch(void* const* d_in, const int* in_sizes, int n_in,
                              void* d_out, int out_size, void* d_ws, size_t ws_size,
                              hipStream_t stream) {
    const float* x   = (const float*)d_in[0];
    const int*   ei  = (const int*)d_in[1];   // [2, E] int32
    const float* W   = (const float*)d_in[2];
    const float* att = (const float*)d_in[3];
    const int* src = ei;
    const int* dst = ei + N_EDGES;
    float* out = (float*)d_out;

    char* ws = (char*)d_ws;
    float* h     = (float*)(ws);
    float* e_arr = (float*)(ws + 12800000);
    float* a_src = (float*)(ws + 16000000);
    float* a_dst = (float*)(ws + 16200000);
    float* m     = (float*)(ws + 16400000);
    float* denom = (float*)(ws + 16600000);

    // 1) init m/denom/out
    {
        const int n = N_NODES * OUT_DIM;
        init_kernel<<<(n + 255) / 256, 256, 0, stream>>>(m, denom, out);
    }
    // 2) h = x @ W  (WMMA, 4 waves/block, one 16-row tile per wave)
    {
        const int tiles = (N_NODES + 15) / 16;      // 3125
        gemm_wmma_kernel<<<(tiles + 3) / 4, 128, 0, stream>>>(x, W, h);
    }
    // 3) per-node attention scalars (wave per node, 8 waves/block)
    node_att_kernel<<<(N_NODES + 7) / 8, 256, 0, stream>>>(h, att, a_src, a_dst);
    // 4) edge logits + segment max
    edge_logit_kernel<<<(N_EDGES + 255) / 256, 256, 0, stream>>>(src, dst, a_src, a_dst, e_arr, m);
    // 5) exp + segment sum
    edge_exp_kernel<<<(N_EDGES + 255) / 256, 256, 0, stream>>>(dst, m, e_arr, denom);
    // 6) weighted scatter-add (16 threads/edge)
    edge_aggregate_kernel<<<(N_EDGES * 16) / 256, 256, 0, stream>>>(src, dst, e_arr, denom, h, out);
    // 7) elu in place
    {
        const int n4 = (N_NODES * OUT_DIM) / 4;
        elu_kernel<<<(n4 + 255) / 256, 256, 0, stream>>>(out);
    }
}